// Motion_Aggregation_Module_6897717477885
// MI455X (gfx1250) — compile-verified
//
#include <hip/hip_runtime.h>
#include <hip/hip_bf16.h>

typedef __attribute__((ext_vector_type(2))) float v2f;
typedef __attribute__((ext_vector_type(8))) float v8f;

union V8 { v8f v; float f[8]; };

// Problem constants (match reference)
constexpr int kB   = 2048;
constexpr int kM   = 5;
constexpr int kS   = 128;
constexpr int kN   = 128;
constexpr int kDIN = 64;
constexpr int kC   = 32;
constexpr int kMID = 16;
constexpr float kEPS = 1e-5f;
constexpr float kSLOPE = 0.2f;

constexpr long  kROWS        = (long)kB * kM * kS;           // 1,310,720
constexpr size_t kCURVES_ELT = (size_t)kROWS * kC;           // 41,943,040 floats
constexpr size_t kFEAT_ELT   = (size_t)kB * kC * kN;         // 8,388,608 floats

#define WMMA_F32(A, Bf, Cc) \
  __builtin_amdgcn_wmma_f32_16x16x4_f32(false, (A), false, (Bf), (short)0, (Cc), false, false)

// A fragment: rows row0..row0+15 (rowbase pre-offset), K chunk k0 (4 wide).
// lane%16 = row, lane/16 picks K pair {k0,k0+1} vs {k0+2,k0+3}.
__device__ __forceinline__ v2f frag_a(const float* rowbase, int ld, int k0, int lane) {
  const float* p = rowbase + (size_t)(lane & 15) * ld + k0 + 2 * (lane >> 4);
  v2f r; r.x = p[0]; r.y = p[1];
  return r;
}
// B fragment from row-major [K][ldn] (B[k][n]): lane%16 = col, lane/16 picks K pair.
__device__ __forceinline__ v2f frag_b(const float* base, int ld, int k0, int col0, int lane) {
  int n = col0 + (lane & 15), kk = k0 + 2 * (lane >> 4);
  v2f r; r.x = base[(size_t)kk * ld + n]; r.y = base[(size_t)(kk + 1) * ld + n];
  return r;
}
// B fragment where B[k][n] = W[n][k], W row-major [cols][ldk].
__device__ __forceinline__ v2f frag_bT(const float* W, int ld, int k0, int col0, int lane) {
  const float* p = W + (size_t)(col0 + (lane & 15)) * ld + k0 + 2 * (lane >> 4);
  v2f r; r.x = p[0]; r.y = p[1];
  return r;
}

// ---------------------------------------------------------------------------
// Kernel 0: zero BN accumulators (64 floats: [0..31]=sum, [32..63]=sumsq)
// ---------------------------------------------------------------------------
__global__ void mam_zero_bn(float* g_bn) {
  if (threadIdx.x < 64) g_bn[threadIdx.x] = 0.0f;
}

// ---------------------------------------------------------------------------
// Kernel 1: curves = x_local @ W_lin^T + b_lin   ([B*M*S,64] x [64,32])
// One wave per 16-row tile; 2 column tiles of 16; K=64 in 16 WMMA chunks.
// curves stored row-major [B,M,S,C].
// ---------------------------------------------------------------------------
__global__ void __launch_bounds__(256)
mam_curves(const float* __restrict__ xl, const float* __restrict__ W_lin,
           const float* __restrict__ b_lin, float* __restrict__ g_curves) {
  const int lane = threadIdx.x & 31;
  const int wave = threadIdx.x >> 5;
  const long tile = (long)blockIdx.x * 8 + wave;
  const long row0 = tile * 16;
  if (row0 >= kROWS) return;
  const float* arow = xl + (size_t)row0 * kDIN;
  const int hl = lane >> 4, ll = lane & 15;

#pragma unroll
  for (int ct = 0; ct < 2; ++ct) {
    v8f acc = {};
#pragma unroll
    for (int k0 = 0; k0 < kDIN; k0 += 4) {
      v2f a  = frag_a(arow, kDIN, k0, lane);
      v2f bf = frag_bT(W_lin, kDIN, k0, ct * 16, lane);  // B[d][c] = W_lin[c][d]
      acc = WMMA_F32(a, bf, acc);
    }
    const int c = ct * 16 + ll;
    const float bias = b_lin[c];
    V8 u; u.v = acc;
#pragma unroll
    for (int v = 0; v < 8; ++v) {
      long r = row0 + v + 8 * hl;
      g_curves[(size_t)r * kC + c] = u.f[v] + bias;
    }
  }
}

// ---------------------------------------------------------------------------
// Kernel 2: per-batch fused correlation module. One 256-thread block per b.
// Dynamic LDS layout (floats):
//   s_att 640 | s_wS 640 | s_wM 640 | s_inter 160 | s_interP 80 | s_vinter 80
//   s_bn 64 | s_intra 4096 | s_intraP 2048 | s_vintra 2048 | s_xlog 2048
//   s_fcat 4096 | s_ast 8*2048   => 33024 floats = 132096 B
// ---------------------------------------------------------------------------
__global__ void __launch_bounds__(256)
mam_batch(const float* __restrict__ xg, const float* __restrict__ g_curves,
          const float* __restrict__ W_att, const float* __restrict__ Wa,
          const float* __restrict__ Wb, const float* __restrict__ Wc,
          const float* __restrict__ Wn, const float* __restrict__ Wl,
          const float* __restrict__ Wd,
          float* __restrict__ g_feat, float* __restrict__ g_bn) {
  extern __shared__ float sm[];
  float* s_att    = sm;
  float* s_wS     = sm + 640;
  float* s_wM     = sm + 1280;
  float* s_inter  = sm + 1920;   // [c*5+m]
  float* s_interP = sm + 2080;   // [o*5+m]
  float* s_vinter = sm + 2160;   // [m*16+p]
  float* s_bn     = sm + 2240;   // 64
  float* s_intra  = sm + 2304;   // [c*128+s]
  float* s_intraP = sm + 6400;   // [o*128+s]
  float* s_vintra = sm + 8448;   // [s*16+p]
  float* s_xlog   = sm + 10496;  // [n*16+o]
  float* s_fcat   = sm + 12544;  // [n*32+k]
  float* s_ast    = sm + 16640;  // per-wave [16*128]

  const int tid  = threadIdx.x;
  const int lane = tid & 31, wave = tid >> 5;
  const int hl = lane >> 4, ll = lane & 15;
  const int b = blockIdx.x;
  const float* cb = g_curves + (size_t)b * (kM * kS * kC);  // [(m*128+s)*32+c]
  const float* xb = xg + (size_t)b * (kN * kC);             // [n*32+c]

  // --- att[m,s] = sum_c W_att[c] * curves[m,s,c] ---
  for (int idx = tid; idx < kM * kS; idx += 256) {
    const float* p = cb + (size_t)idx * kC;
    float a = 0.f;
#pragma unroll
    for (int c = 0; c < kC; ++c) a += p[c] * W_att[c];
    s_att[idx] = a;
  }
  __syncthreads();

  // --- wS: softmax over s per m; wM: softmax over m per s ---
  if (tid < kM) {
    float mx = -1e30f;
    for (int s = 0; s < kS; ++s) mx = fmaxf(mx, s_att[tid * kS + s]);
    float sme = 0.f;
    for (int s = 0; s < kS; ++s) sme += __expf(s_att[tid * kS + s] - mx);
    float inv = 1.f / sme;
    for (int s = 0; s < kS; ++s) s_wS[tid * kS + s] = __expf(s_att[tid * kS + s] - mx) * inv;
  }
  if (tid < kS) {
    float mx = -1e30f;
    for (int m = 0; m < kM; ++m) mx = fmaxf(mx, s_att[m * kS + tid]);
    float sme = 0.f;
    for (int m = 0; m < kM; ++m) sme += __expf(s_att[m * kS + tid] - mx);
    float inv = 1.f / sme;
    for (int m = 0; m < kM; ++m) s_wM[m * kS + tid] = __expf(s_att[m * kS + tid] - mx) * inv;
  }
  __syncthreads();

  // --- intra[c,s] = sum_m curves * wM ; inter[c,m] = sum_s curves * wS ---
  for (int idx = tid; idx < kC * kS; idx += 256) {
    int c = idx >> 7, s = idx & 127;
    float a = 0.f;
#pragma unroll
    for (int m = 0; m < kM; ++m) a += cb[(size_t)(m * kS + s) * kC + c] * s_wM[m * kS + s];
    s_intra[idx] = a;
  }
  for (int idx = tid; idx < kC * kM; idx += 256) {
    int c = idx / kM, m = idx % kM;
    float a = 0.f;
    for (int s = 0; s < kS; ++s) a += cb[(size_t)(m * kS + s) * kC + c] * s_wS[m * kS + s];
    s_inter[c * kM + m] = a;
  }
  if (tid < 64) s_bn[tid] = 0.f;
  __syncthreads();

  // --- intraP = Wb @ intra ; interP = Wa @ inter ---
  for (int idx = tid; idx < kMID * kS; idx += 256) {
    int o = idx >> 7, s = idx & 127;
    float a = 0.f;
#pragma unroll
    for (int c = 0; c < kC; ++c) a += Wb[o * kC + c] * s_intra[c * kS + s];
    s_intraP[idx] = a;
  }
  if (tid < kMID * kM) {
    int o = tid / kM, m = tid % kM;
    float a = 0.f;
#pragma unroll
    for (int c = 0; c < kC; ++c) a += Wa[o * kC + c] * s_inter[c * kM + m];
    s_interP[o * kM + m] = a;
  }
  __syncthreads();

  // --- v_intra[s,p] = sum_o Wl[p,o]*intraP[o,s] ; v_inter[m,p] similarly ---
  for (int idx = tid; idx < kS * kMID; idx += 256) {
    int s = idx >> 4, p = idx & 15;
    float a = 0.f;
#pragma unroll
    for (int o = 0; o < kMID; ++o) a += Wl[p * kMID + o] * s_intraP[o * kS + s];
    s_vintra[idx] = a;
  }
  if (tid < kM * kMID) {
    int m = tid >> 4, p = tid & 15;
    float a = 0.f;
#pragma unroll
    for (int o = 0; o < kMID; ++o) a += Wn[p * kMID + o] * s_interP[o * kM + m];
    s_vinter[tid] = a;
  }

  // --- x_logits[n,o] = sum_c x[n,c]*Wc[o,c]  (WMMA, wave handles 16 rows) ---
  {
    const int n0 = wave * 16;
    v8f acc = {};
#pragma unroll
    for (int k0 = 0; k0 < kC; k0 += 4) {
      v2f a  = frag_a(xb + (size_t)n0 * kC, kC, k0, lane);
      v2f bf = frag_bT(Wc, kC, k0, 0, lane);  // B[c][o] = Wc[o][c]
      acc = WMMA_F32(a, bf, acc);
    }
    V8 u; u.v = acc;
#pragma unroll
    for (int v = 0; v < 8; ++v) s_xlog[(n0 + v + 8 * hl) * kMID + ll] = u.f[v];
  }
  __syncthreads();

  // --- f_inter (scalar path, tiny M=5) -> fcat[n, 0..15] ---
  for (int idx = tid; idx < kN * kMID; idx += 256) {
    int n = idx >> 4, p = idx & 15;
    float lg[kM];
    float mx = -1e30f;
#pragma unroll
    for (int m = 0; m < kM; ++m) {
      float a = 0.f;
#pragma unroll
      for (int o = 0; o < kMID; ++o) a += s_xlog[n * kMID + o] * s_interP[o * kM + m];
      lg[m] = a; mx = fmaxf(mx, a);
    }
    float sme = 0.f;
#pragma unroll
    for (int m = 0; m < kM; ++m) { lg[m] = __expf(lg[m] - mx); sme += lg[m]; }
    float inv = 1.f / sme, f = 0.f;
#pragma unroll
    for (int m = 0; m < kM; ++m) f += lg[m] * inv * s_vinter[m * kMID + p];
    s_fcat[n * kC + p] = f;
  }

  // --- a_intra logits (WMMA) + register softmax + f_intra (WMMA) ---
  {
    const int n0 = wave * 16;
    V8 L[8];
#pragma unroll
    for (int j = 0; j < 8; ++j) {
      v8f acc = {};
#pragma unroll
      for (int k0 = 0; k0 < kMID; k0 += 4) {
        v2f a  = frag_a(s_xlog + n0 * kMID, kMID, k0, lane);
        v2f bf = frag_b(s_intraP, kS, k0, j * 16, lane);  // B[o][s]
        acc = WMMA_F32(a, bf, acc);
      }
      L[j].v = acc;
    }
    // Row softmax across 128 cols: per-vgpr reduce across j then 16-lane half.
#pragma unroll
    for (int v = 0; v < 8; ++v) {
      float mx = -1e30f;
#pragma unroll
      for (int j = 0; j < 8; ++j) mx = fmaxf(mx, L[j].f[v]);
#pragma unroll
      for (int msk = 1; msk < 16; msk <<= 1) mx = fmaxf(mx, __shfl_xor(mx, msk, 32));
      float sme = 0.f;
#pragma unroll
      for (int j = 0; j < 8; ++j) { float e = __expf(L[j].f[v] - mx); L[j].f[v] = e; sme += e; }
#pragma unroll
      for (int msk = 1; msk < 16; msk <<= 1) sme += __shfl_xor(sme, msk, 32);
      float inv = 1.f / sme;
#pragma unroll
      for (int j = 0; j < 8; ++j) L[j].f[v] *= inv;
    }
    // Stage probabilities [16][128] to per-wave LDS for A-fragment re-layout.
    float* ast = s_ast + wave * (16 * kS);
#pragma unroll
    for (int v = 0; v < 8; ++v)
#pragma unroll
      for (int j = 0; j < 8; ++j) ast[(v + 8 * hl) * kS + j * 16 + ll] = L[j].f[v];
    __syncthreads();  // uniform across block

    v8f F = {};
#pragma unroll
    for (int k0 = 0; k0 < kS; k0 += 4) {
      v2f a  = frag_a(ast, kS, k0, lane);
      v2f bf = frag_b(s_vintra, kMID, k0, 0, lane);  // B[s][p]
      F = WMMA_F32(a, bf, F);
    }
    V8 u; u.v = F;
#pragma unroll
    for (int v = 0; v < 8; ++v) s_fcat[(n0 + v + 8 * hl) * kC + kMID + ll] = u.f[v];
  }
  __syncthreads();

  // --- feat[c,n] = Wd @ fcat^T (WMMA) + BN partial sums ---
  {
    const int n0 = wave * 16;
#pragma unroll
    for (int ct = 0; ct < 2; ++ct) {
      v8f acc = {};
#pragma unroll
      for (int k0 = 0; k0 < kC; k0 += 4) {
        v2f a  = frag_a(s_fcat + n0 * kC, kC, k0, lane);
        v2f bf = frag_bT(Wd, kC, k0, ct * 16, lane);  // B[k][c] = Wd[c][k]
        acc = WMMA_F32(a, bf, acc);
      }
      const int c = ct * 16 + ll;
      V8 u; u.v = acc;
      float ps = 0.f, pq = 0.f;
#pragma unroll
      for (int v = 0; v < 8; ++v) {
        float val = u.f[v];
        g_feat[((size_t)b * kC + c) * kN + n0 + v + 8 * hl] = val;
        ps += val; pq += val * val;
      }
      atomicAdd(&s_bn[c], ps);
      atomicAdd(&s_bn[kC + c], pq);
    }
  }
  __syncthreads();
  if (tid < 64) atomicAdd(&g_bn[tid], s_bn[tid]);
}

// ---------------------------------------------------------------------------
// Kernel 3: BN (batch stats) + residual + leaky ReLU, output [B,N,C]
// ---------------------------------------------------------------------------
__global__ void __launch_bounds__(256)
mam_finish(const float* __restrict__ xg, const float* __restrict__ g_feat,
           const float* __restrict__ g_bn, const float* __restrict__ gamma,
           const float* __restrict__ beta, float* __restrict__ out) {
  size_t idx = (size_t)blockIdx.x * 256 + threadIdx.x;
  if (idx >= (size_t)kB * kN * kC) return;
  int c = (int)(idx & (kC - 1));
  size_t bn = idx >> 5;
  int n = (int)(bn & (kN - 1));
  size_t b = bn >> 7;
  const float cnt = (float)((size_t)kB * kN);
  float mean = g_bn[c] / cnt;
  float var  = g_bn[kC + c] / cnt - mean * mean;
  float scl  = gamma[c] * rsqrtf(var + kEPS);
  float f = (g_feat[(b * kC + c) * kN + n] - mean) * scl + beta[c];
  float val = xg[idx] + f;
  out[idx] = val >= 0.f ? val : kSLOPE * val;
}

// ---------------------------------------------------------------------------
extern "C" void kernel_launch(void* const* d_in, const int* in_sizes, int n_in,
                              void* d_out, int out_size, void* d_ws, size_t ws_size,
                              hipStream_t stream) {
  (void)in_sizes; (void)n_in; (void)out_size; (void)ws_size;
  const float* xg    = (const float*)d_in[0];
  const float* xl    = (const float*)d_in[1];
  const float* W_lin = (const float*)d_in[2];
  const float* b_lin = (const float*)d_in[3];
  const float* W_att = (const float*)d_in[4];
  const float* Wa    = (const float*)d_in[5];
  const float* Wb    = (const float*)d_in[6];
  const float* Wc    = (const float*)d_in[7];
  const float* Wn    = (const float*)d_in[8];
  const float* Wl    = (const float*)d_in[9];
  const float* Wd    = (const float*)d_in[10];
  const float* gamma = (const float*)d_in[11];
  const float* beta  = (const float*)d_in[12];

  float* ws       = (float*)d_ws;
  float* g_curves = ws;
  float* g_feat   = ws + kCURVES_ELT;
  float* g_bn     = ws + kCURVES_ELT + kFEAT_ELT;
  float* out      = (float*)d_out;

  mam_zero_bn<<<1, 64, 0, stream>>>(g_bn);

  // B*M*S rows / 16 per tile / 8 tiles per block = 10240 blocks
  mam_curves<<<(int)(kROWS / 16 / 8), 256, 0, stream>>>(xl, W_lin, b_lin, g_curves);

  const size_t smbytes = 33024 * sizeof(float);  // ~129 KB dynamic LDS
  mam_batch<<<kB, 256, smbytes, stream>>>(xg, g_curves, W_att, Wa, Wb, Wc, Wn, Wl, Wd,
                                          g_feat, g_bn);

  mam_finish<<<(int)(((size_t)kB * kN * kC) / 256), 256, 0, stream>>>(
      xg, g_feat, g_bn, gamma, beta, out);
}